// LinearEquivariant_57939108823731
// MI455X (gfx1250) — compile-verified
//
#include <hip/hip_runtime.h>

// LinearEquivariant on MI455X (gfx1250):
//   out = (x @ alpha)/128 + row_bias[b, o]
//   row_bias = (x_sum @ beta + sum_c bias)/128
// Main GEMM: V_WMMA_F32_16X16X4_F32 with double-buffered
// GLOBAL_LOAD_ASYNC_TO_LDS_B128 staging of x tiles.

typedef float v2f __attribute__((ext_vector_type(2)));
typedef float v8f __attribute__((ext_vector_type(8)));

#define NB 8
#define ND 32768
#define NC 128
#define NO 128
#define SLABS 16          // slabs per batch for colsum reduction
#define SLAB_ROWS (ND / SLABS)
#define TILES_PER_WG 8
#define XT_STRIDE 132     // 128 + 4 pad: conflict-free ds_load_b64 A-fragments
#define BUF_BYTES (16 * XT_STRIDE * 4)       // 8448 B per x-tile buffer
#define CHUNK1_BYTES (8 * XT_STRIDE * 4)     // second b128 chunk: +8 rows

// ---- generic -> LDS byte-offset (ptrtoint of addrspace(3) pointer) --------
__device__ __forceinline__ unsigned lds_off_of(const void* p) {
    return (unsigned)(unsigned long long)
        (__attribute__((address_space(3))) const void*)p;
}

// ---- CDNA5 async copy: 16B global -> LDS per lane, tracked by ASYNCcnt ----
__device__ __forceinline__ void async_load_b128(unsigned lds_off, const float* gaddr) {
    asm volatile("global_load_async_to_lds_b128 %0, %1, off"
                 :: "v"(lds_off), "v"(gaddr)
                 : "memory");
}
__device__ __forceinline__ void wait_asynccnt0() {
    asm volatile("s_wait_asynccnt 0x0" ::: "memory");
}

// ---------------- Kernel 1: deterministic column sums (partial per slab) ---
__global__ __launch_bounds__(256)
void colsum_kernel(const float* __restrict__ x, float* __restrict__ partials)
{
    __shared__ float red[256];
    const int blk  = blockIdx.x;            // 0 .. NB*SLABS-1
    const int b    = blk >> 4;
    const int slab = blk & (SLABS - 1);
    const int tid  = threadIdx.x;
    const int c    = tid & 127;
    const int dd   = tid >> 7;              // 0 or 1

    const float* p = x + ((long)b * ND + (long)slab * SLAB_ROWS + dd) * NC + c;
    float acc = 0.f;
#pragma unroll 8
    for (int i = 0; i < SLAB_ROWS / 2; ++i)
        acc += p[(long)2 * i * NC];

    red[tid] = acc;
    __syncthreads();
    if (tid < 128)
        partials[blk * NC + c] = red[tid] + red[tid + 128];
}

// ---------------- Kernel 2: row_bias[b][o] -------------------------------
__global__ __launch_bounds__(128)
void rowbias_kernel(const float* __restrict__ partials,
                    const float* __restrict__ beta,
                    const float* __restrict__ bias,
                    float* __restrict__ row_bias)
{
    __shared__ float ssum[NC];
    const int b = blockIdx.x;
    const int o = threadIdx.x;

    float s = 0.f;
#pragma unroll
    for (int j = 0; j < SLABS; ++j)
        s += partials[(b * SLABS + j) * NC + o];   // index doubles as c here
    ssum[o] = s;
    __syncthreads();

    float rb = 0.f, bm = 0.f;
#pragma unroll 8
    for (int ci = 0; ci < NC; ++ci) {
        rb = fmaf(ssum[ci], beta[ci * NO + o], rb);
        bm += bias[ci * NO + o];
    }
    row_bias[b * NO + o] = (rb + bm) * 0.0078125f;  // /128
}

// ---------------- Kernel 3: WMMA GEMM + async double-buffered staging ------
// Block = 8 waves. Wave w owns N-tile [16w, 16w+16). B-fragments of alpha
// (all 32 k-steps) preloaded into 64 VGPRs, reused across TILES_PER_WG
// M-tiles. x tiles staged via GLOBAL_LOAD_ASYNC_TO_LDS_B128 into a
// double-buffered LDS tile (stride 132 -> bank-conflict-free b64 A reads).
__global__ __launch_bounds__(256)
void gemm_kernel(const float* __restrict__ x,
                 const float* __restrict__ alpha,
                 const float* __restrict__ row_bias,
                 float* __restrict__ out)
{
    __shared__ float xtile[2][16 * XT_STRIDE];

    const int tid  = threadIdx.x;
    const int wave = tid >> 5;
    const int lane = tid & 31;
    const int l15  = lane & 15;
    const int half = lane >> 4;          // 0 or 1
    const int n0   = wave * 16;

    // Staging: thread -> (row = tid>>5, col4 = tid&31) and the chunk +8 rows.
    const unsigned lds0 = lds_off_of(&xtile[0][(tid >> 5) * XT_STRIDE + (tid & 31) * 4]);

    // Preload B fragments: bfrag[s] covers K = 4s..4s+3 of alpha[:, n0+l15].
    // ISA B layout: v0 = row K (lanes0-15) / K+2 (lanes16-31), v1 = K+1 / K+3.
    v2f bfrag[32];
#pragma unroll
    for (int s = 0; s < 32; ++s) {
        const int k = 4 * s + 2 * half;
        bfrag[s].x = alpha[(k + 0) * NO + n0 + l15];
        bfrag[s].y = alpha[(k + 1) * NO + n0 + l15];
    }

    const int tm_base = blockIdx.x * TILES_PER_WG;       // 2048 tiles/batch, 8|2048
    const int b       = tm_base >> 11;
    const float rb    = row_bias[b * NO + n0 + l15];

    // Running pointers (incremented per m-tile; no per-iteration re-derivation).
    const float* g  = x + (long)tm_base * 16 * NC + (tid >> 5) * NC + (tid & 31) * 4;
    float*       op = out + ((long)tm_base * 16 + half * 8) * NO + n0 + l15;
    const float* aBase0 = &xtile[0][l15 * XT_STRIDE + 2 * half];

    // Prologue: async-fill buffer 0 with tile 0.
    async_load_b128(lds0, g);
    async_load_b128(lds0 + CHUNK1_BYTES, g + 8 * NC);
    g += 16 * NC;
    wait_asynccnt0();
    __syncthreads();

    for (int t = 0; t < TILES_PER_WG; ++t) {
        const int cur = t & 1;

        // Kick off async fill of the other buffer with tile t+1.
        if (t + 1 < TILES_PER_WG) {
            const unsigned dst = lds0 + (unsigned)((cur ^ 1) * BUF_BYTES);
            async_load_b128(dst, g);
            async_load_b128(dst + CHUNK1_BYTES, g + 8 * NC);
            g += 16 * NC;
        }

        // Compute on current buffer: 32 k-steps of f32 WMMA, with an explicit
        // two-deep A-fragment pipeline (load s+1 while WMMA s executes).
        const float* ap = aBase0 + cur * (16 * XT_STRIDE);
        v2f afrag[2];
        afrag[0].x = ap[0];
        afrag[0].y = ap[1];

        v8f acc = {};
#pragma unroll
        for (int s = 0; s < 32; ++s) {
            if (s + 1 < 32) {
                afrag[(s + 1) & 1].x = ap[4 * (s + 1)];
                afrag[(s + 1) & 1].y = ap[4 * (s + 1) + 1];
            }
            acc = __builtin_amdgcn_wmma_f32_16x16x4_f32(
                false, afrag[s & 1], false, bfrag[s], (short)0, acc, false, false);
        }

        // Epilogue: C/D layout -> VGPR i holds (M = i + 8*half, N = l15).
        // Non-temporal stores: keep the 134MB output out of L2 so x stays hot.
#pragma unroll
        for (int i = 0; i < 8; ++i)
            __builtin_nontemporal_store(fmaf(acc[i], 0.0078125f, rb), op + i * NO);
        op += 16 * NO;

        // Next buffer must be resident in LDS for everyone before reuse.
        wait_asynccnt0();
        __syncthreads();
    }
}

// ---------------- Launch ---------------------------------------------------
extern "C" void kernel_launch(void* const* d_in, const int* in_sizes, int n_in,
                              void* d_out, int out_size, void* d_ws, size_t ws_size,
                              hipStream_t stream) {
    const float* x     = (const float*)d_in[0];
    const float* alpha = (const float*)d_in[1];
    const float* beta  = (const float*)d_in[2];
    const float* bias  = (const float*)d_in[3];
    float* out = (float*)d_out;

    float* partials = (float*)d_ws;                 // NB*SLABS*NC = 16384 floats
    float* row_bias = partials + NB * SLABS * NC;   // NB*NO = 1024 floats

    colsum_kernel <<<NB * SLABS, 256, 0, stream>>>(x, partials);
    rowbias_kernel<<<NB, 128, 0, stream>>>(partials, beta, bias, row_bias);

    const int m_tiles = (NB * ND) / 16;             // 16384
    gemm_kernel   <<<m_tiles / TILES_PER_WG, 256, 0, stream>>>(x, alpha, row_bias, out);
}